// VAE_75806172774966
// MI455X (gfx1250) — compile-verified
//
#include <hip/hip_runtime.h>
#include <hip/hip_bf16.h>

typedef float v2f __attribute__((ext_vector_type(2)));
typedef float v8f __attribute__((ext_vector_type(8)));

#define BB 64
#define LLN 80
#define DDN 10
#define HHN 32
#define NSTEP 100

__device__ __forceinline__ v2f mk2(float x, float y) { v2f r; r.x = x; r.y = y; return r; }

__device__ __forceinline__ v8f wmma4(v2f a, v2f b, v8f c) {
    // V_WMMA_F32_16X16X4_F32: full-precision fp32 matrix op, wave32
    return __builtin_amdgcn_wmma_f32_16x16x4_f32(false, a, false, b, (short)0, c, false, false);
}

__device__ __forceinline__ unsigned rotl32(unsigned x, unsigned r) {
    return (x << r) | (x >> (32u - r));
}

// JAX threefry2x32, 20 rounds
__device__ __forceinline__ void threefry2x32(unsigned k0, unsigned k1,
                                             unsigned x0, unsigned x1,
                                             unsigned& o0, unsigned& o1) {
    unsigned ks0 = k0, ks1 = k1, ks2 = k0 ^ k1 ^ 0x1BD11BDAu;
    unsigned ks[3] = {ks0, ks1, ks2};
    const unsigned rot[2][4] = {{13u, 15u, 26u, 6u}, {17u, 29u, 16u, 24u}};
    x0 += ks[0]; x1 += ks[1];
#pragma unroll
    for (int i = 0; i < 5; ++i) {
        const unsigned* r = rot[i & 1];
#pragma unroll
        for (int j = 0; j < 4; ++j) { x0 += x1; x1 = rotl32(x1, r[j]); x1 ^= x0; }
        x0 += ks[(i + 1) % 3];
        x1 += ks[(i + 2) % 3] + (unsigned)(i + 1);
    }
    o0 = x0; o1 = x1;
}

// ---------------- Kernel A: basek[b][j] = b1[j] + sum_k zmask[b,k]*W1[k,j] --------------
// 8 blocks x 32 threads; block = (tt, h): 16 rows of b, 16 cols of j. K=800 via 200 chained WMMAs.
__global__ __launch_bounds__(32) void basek_kernel(const float* __restrict__ z_mean,
                                                   const float* __restrict__ W1,
                                                   const float* __restrict__ b1,
                                                   float* __restrict__ basek) {
    const int blk = blockIdx.x;
    const int tt = blk & 3;      // b-tile
    const int h = blk >> 2;      // j-half
    const int l = threadIdx.x;
    const int mn = l & 15;
    const int half = l >> 4;
    const int mbase = half * 8;
    const int kb = half * 2;

    v8f c;
#pragma unroll
    for (int p = 0; p < 8; ++p) c[p] = 0.f;

    const int b = 16 * tt + mn;  // A-row
    for (int ch = 0; ch < 200; ++ch) {
        const int k0 = 4 * ch + kb;
        const int k1 = k0 + 1;
        // zmask[b][k] = z_mean[b, k/10, k%10] kept iff (k%10) < (k/10)  (band * off-diagonal)
        float a0 = ((k0 % 10) < (k0 / 10)) ? z_mean[b * 800 + k0] : 0.f;
        float a1 = ((k1 % 10) < (k1 / 10)) ? z_mean[b * 800 + k1] : 0.f;
        float w0 = W1[k0 * 32 + 16 * h + mn];
        float w1 = W1[k1 * 32 + 16 * h + mn];
        c = wmma4(mk2(a0, a1), mk2(w0, w1), c);
    }
    const float bias = b1[16 * h + mn];
#pragma unroll
    for (int p = 0; p < 8; ++p)
        basek[(16 * tt + mbase + p) * 32 + 16 * h + mn] = c[p] + bias;
}

// ---------------- Kernel B: 100-step Euler-Maruyama with WMMA score net --------------
// 320 blocks x 32 threads (one wave per 16-row tile). block = (i, tt): rows b = 16tt..16tt+15,
// all sharing step slot i, so they share the W1[i*10 : i*10+10, :] slice.
__global__ __launch_bounds__(32) void diff_step_kernel(const float* __restrict__ z_mean,
                                                       const float* __restrict__ W1,
                                                       const float* __restrict__ W2,
                                                       const float* __restrict__ b2,
                                                       const float* __restrict__ basek,
                                                       float* __restrict__ partials) {
    __shared__ float accLDS[16][16];  // C-layout staging for GEMM1 A operand
    __shared__ float preLDS[16][32];  // relu(pre) staging for GEMM2 A operand

    const int blk = blockIdx.x;
    const int i = blk % 80;
    const int tt = blk / 80;
    const int l = threadIdx.x;
    const int mn = l & 15;      // C: column n; A/B frags: row/col index
    const int half = l >> 4;
    const int mbase = half * 8; // C rows owned by this lane
    const int kb = half * 2;    // K base within a K=4 chunk for A/B layouts

    // --- W1 slice B fragments: K=12 (pad from 10), N=32 as two halves, 3 K-chunks ---
    v2f bw1[2][3];
#pragma unroll
    for (int h = 0; h < 2; ++h)
#pragma unroll
        for (int c = 0; c < 3; ++c) {
            const int kk0 = 4 * c + kb;
            float v0 = (kk0 < 10) ? W1[(i * 10 + kk0) * 32 + 16 * h + mn] : 0.f;
            float v1 = (kk0 + 1 < 10) ? W1[(i * 10 + kk0 + 1) * 32 + 16 * h + mn] : 0.f;
            bw1[h][c] = mk2(v0, v1);
        }
    // --- W2 B fragments: K=32 -> 8 chunks, N=16 (cols >=10 zero-padded) ---
    v2f bw2[8];
#pragma unroll
    for (int c = 0; c < 8; ++c) {
        const int kk0 = 4 * c + kb;
        float v0 = 0.f, v1 = 0.f;
        if (mn < 10) { v0 = W2[kk0 * 10 + mn]; v1 = W2[(kk0 + 1) * 10 + mn]; }
        bw2[c] = mk2(v0, v1);
    }
    // time-column row of W1 and biases (broadcast along M in C layout)
    const float w1t0 = W1[800 * 32 + mn];
    const float w1t1 = W1[800 * 32 + 16 + mn];
    const float b2v = (mn < 10) ? b2[mn] : 0.f;

    // basek C fragments (constant across steps)
    v8f baseF[2];
#pragma unroll
    for (int h = 0; h < 2; ++h)
#pragma unroll
        for (int p = 0; p < 8; ++p)
            baseF[h][p] = basek[(16 * tt + mbase + p) * 32 + 16 * h + mn];

    v8f acc;
#pragma unroll
    for (int p = 0; p < 8; ++p) acc[p] = 0.f;

    for (int s = 0; s < NSTEP; ++s) {
        const float t = (float)s * 0.01f;

        // stage acc (C layout) into LDS, reload as A-layout K=4 chunks
#pragma unroll
        for (int p = 0; p < 8; ++p) accLDS[mbase + p][mn] = acc[p];
        __syncthreads();
        v2f a1[3];
#pragma unroll
        for (int c = 0; c < 3; ++c)
            a1[c] = mk2(accLDS[mn][4 * c + kb], accLDS[mn][4 * c + kb + 1]);

        // GEMM1: pre = basek + t*W1[800,:] + acc @ W1slice ; relu
        v8f pre[2];
#pragma unroll
        for (int h = 0; h < 2; ++h) {
            v8f c0;
            const float tw = t * (h ? w1t1 : w1t0);
#pragma unroll
            for (int p = 0; p < 8; ++p) c0[p] = baseF[h][p] + tw;
            c0 = wmma4(a1[0], bw1[h][0], c0);
            c0 = wmma4(a1[1], bw1[h][1], c0);
            c0 = wmma4(a1[2], bw1[h][2], c0);
#pragma unroll
            for (int p = 0; p < 8; ++p) c0[p] = fmaxf(c0[p], 0.f);
            pre[h] = c0;
        }

        // stage relu(pre), reload as A-layout chunks for GEMM2
        __syncthreads();
#pragma unroll
        for (int h = 0; h < 2; ++h)
#pragma unroll
            for (int p = 0; p < 8; ++p) preLDS[mbase + p][16 * h + mn] = pre[h][p];
        __syncthreads();

        v8f sc;
#pragma unroll
        for (int p = 0; p < 8; ++p) sc[p] = b2v;
#pragma unroll
        for (int c = 0; c < 8; ++c) {
            v2f a2 = mk2(preLDS[mn][4 * c + kb], preLDS[mn][4 * c + kb + 1]);
            sc = wmma4(a2, bw2[c], sc);
        }

        // acc += 0.5*score + dW  (dW: JAX threefry normal, key = fold_in(key(42), s))
        unsigned ka, kk;
        threefry2x32(0u, 42u, 0u, (unsigned)s, ka, kk);
#pragma unroll
        for (int p = 0; p < 8; ++p) {
            float dw = 0.f;
            if (mn < 10) {
                const int b = 16 * tt + mbase + p;
                const unsigned idx = (unsigned)((b * 80 + i) * 10 + mn);  // flat (BL*D) index
                const unsigned j = (idx < 25600u) ? idx : idx - 25600u;
                unsigned o0, o1;
                threefry2x32(ka, kk, j, j + 25600u, o0, o1);
                const unsigned bits = (idx < 25600u) ? o0 : o1;
                const float f = __uint_as_float((bits >> 9) | 0x3f800000u) - 1.0f;
                const float lo = -0.99999994f;  // nextafter(-1, 0)
                const float u = fmaxf(lo, f * (1.0f - lo) + lo);
                dw = 1.4142135623730951f * erfinvf(u) * 0.1f;  // sqrt(2)*erfinv(u)*sqrt(dt)
            }
            acc[p] += 0.5f * sc[p] + dw;
        }
        __syncthreads();
    }

    // per-tile |target - acc| partial (target = z_mean[b,i,i] at d==i, else 0)
    float part = 0.f;
    if (mn < 10) {
#pragma unroll
        for (int p = 0; p < 8; ++p) {
            const int b = 16 * tt + mbase + p;
            const float tgt = (mn == i) ? z_mean[b * 800 + i * 10 + mn] : 0.f;
            part += fabsf(tgt - acc[p]);
        }
    }
#pragma unroll
    for (int off = 16; off > 0; off >>= 1)
        part += __shfl_xor(part, off, 32);
    if (l == 0) partials[blk] = part;
}

// ---------------- Kernel C: deterministic final reduction --------------
__global__ __launch_bounds__(256) void reduce_kernel(const float* __restrict__ z_mean,
                                                     const float* __restrict__ partials,
                                                     float* __restrict__ out) {
    __shared__ float sm[256];
    const int tid = threadIdx.x;
    float s = 0.f;
    for (int idx = tid; idx < 320; idx += 256) s += partials[idx];
    // closed-form off-slot diagonal term: (L-1) * sum_b sum_{l<10} |z_mean[b,l,l]|
    for (int idx = tid; idx < 640; idx += 256) {
        const int b = idx / 10, ll = idx % 10;
        s += 79.0f * fabsf(z_mean[b * 800 + ll * 10 + ll]);
    }
    sm[tid] = s;
    __syncthreads();
    for (int off = 128; off > 0; off >>= 1) {
        if (tid < off) sm[tid] += sm[tid + off];
        __syncthreads();
    }
    if (tid == 0) out[0] = sm[0] / 4096000.0f;  // mean over B*L*L*D
}

extern "C" void kernel_launch(void* const* d_in, const int* in_sizes, int n_in,
                              void* d_out, int out_size, void* d_ws, size_t ws_size,
                              hipStream_t stream) {
    const float* z_mean = (const float*)d_in[0];
    // d_in[1] = z_log_var: unused (faithful to reference)
    const float* W1 = (const float*)d_in[2];
    const float* b1 = (const float*)d_in[3];
    const float* W2 = (const float*)d_in[4];
    const float* b2 = (const float*)d_in[5];

    float* basek = (float*)d_ws;            // 64*32 floats
    float* partials = basek + 64 * 32;      // 320 floats

    basek_kernel<<<8, 32, 0, stream>>>(z_mean, W1, b1, basek);
    diff_step_kernel<<<320, 32, 0, stream>>>(z_mean, W1, W2, b2, basek, partials);
    reduce_kernel<<<1, 256, 0, stream>>>(z_mean, partials, (float*)d_out);
}